// Pat_11536282157369
// MI455X (gfx1250) — compile-verified
//
#include <hip/hip_runtime.h>
#include <math.h>

// ---------------------------------------------------------------------------
// Problem constants (from reference):
//   x: (16, 3, 512, 512) fp32, K=16 -> 32x32 tiles, ch = c*256 + ki*16 + kj
//   pooled: (16, 768); h: (16, 1536); s: (16, 768)
//   out = x^2 * s[b, ch]
// ---------------------------------------------------------------------------

typedef __attribute__((ext_vector_type(2))) float v2f;
typedef __attribute__((ext_vector_type(4))) float v4f;
typedef __attribute__((ext_vector_type(8))) float v8f;

#define NB   16          // batch
#define NC   3           // channels
#define HW   512         // H = W = 512
#define TP   32          // tile (pool window) = 32x32
#define CKK  768         // c*K*K
#define CHID 1536        // hidden

// ---------------------------------------------------------------------------
// Kernel 1: 32x32 tile mean pooling. One wave32 per tile, 8 waves per block.
// pooled[b*768 + c*256 + ki*16 + kj]
// ---------------------------------------------------------------------------
__global__ void se_pool(const float* __restrict__ x, float* __restrict__ pooled) {
    const int lane = threadIdx.x & 31;
    const int wave = threadIdx.x >> 5;
    const int tile = blockIdx.x * 8 + wave;          // 0 .. 12287
    // tile = ((b*3 + c)*16 + ki)*16 + kj
    const int kj = tile & 15;
    const int ki = (tile >> 4) & 15;
    const int bc = tile >> 8;                        // b*3 + c
    const int c  = bc % NC;
    const int b  = bc / NC;

    // lane = hi: each lane sums one 32-float contiguous row of the tile
    const float* row = x + (((size_t)bc * HW) + (size_t)(ki * TP + lane)) * HW
                         + (size_t)(kj * TP);
    const v4f* r4 = (const v4f*)row;                 // 128B-aligned
    float sum = 0.f;
#pragma unroll
    for (int i = 0; i < 8; ++i) {
        v4f v = r4[i];
        sum += (v.x + v.y) + (v.z + v.w);
    }
    // wave32 butterfly reduction
#pragma unroll
    for (int off = 16; off > 0; off >>= 1)
        sum += __shfl_xor(sum, off, 32);

    if (lane == 0)
        pooled[(size_t)b * CKK + c * 256 + ki * 16 + kj] = sum * (1.0f / 1024.0f);
}

// ---------------------------------------------------------------------------
// WMMA fragment loaders for V_WMMA_F32_16X16X4_F32 (wave32).
// A (16x4):  lane L -> row m = L%16 ; VGPR v holds K = (L/16)*2 + v   (ISA 7.12.2)
// B (4x16):  lane L -> col n = L%16 ; VGPR v holds K = (L/16)*2 + v   (row-striped)
// C/D(16x16):lane L -> col n = L%16 ; VGPR v holds row m = v + (L>=16 ? 8 : 0)
// ---------------------------------------------------------------------------

// GEMM1: h(16x1536) = relu( pooled(16x768) @ W2^T ), W2 is (1536,768) row-major
//   h[m, n] = sum_k pooled[m*768+k] * W2[n*768+k]
// One wave per 16-wide N tile (96 tiles); 8 waves/block -> 12 blocks.
__global__ void se_gemm1(const float* __restrict__ pooled,
                         const float* __restrict__ W2,
                         float* __restrict__ h) {
    const int lane  = threadIdx.x & 31;
    const int wave  = threadIdx.x >> 5;
    const int ntile = blockIdx.x * 8 + wave;         // 0 .. 95
    const int m     = lane & 15;                     // A row / B col / D col
    const int kh    = (lane >> 4) * 2;               // K sub-offset per half-wave

    const float* aBase = pooled + (size_t)m * CKK + kh;
    const float* bBase = W2 + (size_t)(ntile * 16 + m) * CKK + kh;

    v8f acc = {0.f, 0.f, 0.f, 0.f, 0.f, 0.f, 0.f, 0.f};
#pragma unroll 4
    for (int ks = 0; ks < CKK; ks += 4) {
        v2f a = *(const v2f*)(aBase + ks);
        v2f bb = *(const v2f*)(bBase + ks);
        acc = __builtin_amdgcn_wmma_f32_16x16x4_f32(
            /*neg_a=*/false, a, /*neg_b=*/false, bb,
            /*c_mod=*/(short)0, acc, /*reuse_a=*/false, /*reuse_b=*/false);
    }

    const int rowAdd = (lane < 16) ? 0 : 8;
    const int col    = ntile * 16 + m;
#pragma unroll
    for (int v = 0; v < 8; ++v) {
        float val = acc[v];
        val = val > 0.f ? val : 0.f;                 // ReLU
        h[(size_t)(rowAdd + v) * CHID + col] = val;
    }
}

// GEMM2: s(16x768) = sigmoid( h(16x1536) @ W3^T + b3 ), W3 is (768,1536)
// One wave per 16-wide N tile (48 tiles); 8 waves/block -> 6 blocks.
__global__ void se_gemm2(const float* __restrict__ h,
                         const float* __restrict__ W3,
                         const float* __restrict__ b3,
                         float* __restrict__ s) {
    const int lane  = threadIdx.x & 31;
    const int wave  = threadIdx.x >> 5;
    const int ntile = blockIdx.x * 8 + wave;         // 0 .. 47
    const int m     = lane & 15;
    const int kh    = (lane >> 4) * 2;

    const float* aBase = h + (size_t)m * CHID + kh;
    const float* bBase = W3 + (size_t)(ntile * 16 + m) * CHID + kh;

    v8f acc = {0.f, 0.f, 0.f, 0.f, 0.f, 0.f, 0.f, 0.f};
#pragma unroll 4
    for (int ks = 0; ks < CHID; ks += 4) {
        v2f a = *(const v2f*)(aBase + ks);
        v2f bb = *(const v2f*)(bBase + ks);
        acc = __builtin_amdgcn_wmma_f32_16x16x4_f32(
            false, a, false, bb, (short)0, acc, false, false);
    }

    const int rowAdd = (lane < 16) ? 0 : 8;
    const int col    = ntile * 16 + m;
    const float bias = b3[col];
#pragma unroll
    for (int v = 0; v < 8; ++v) {
        float val = acc[v] + bias;
        val = 1.0f / (1.0f + __expf(-val));          // sigmoid
        s[(size_t)(rowAdd + v) * CKK + col] = val;
    }
}

// ---------------------------------------------------------------------------
// Kernel 4: out = x^2 * s[b, ch].  float4-vectorized; this is the bandwidth-
// critical kernel (~100 MB HBM traffic; x re-read should hit the 192MB L2).
// ---------------------------------------------------------------------------
__global__ void se_modulate(const float* __restrict__ x,
                            const float* __restrict__ s,
                            float* __restrict__ out) {
    const size_t i4  = (size_t)blockIdx.x * blockDim.x + threadIdx.x; // float4 idx
    const size_t idx = i4 * 4;
    const int    w   = (int)(idx & (HW - 1));        // W coordinate (aligned mod 4)
    const size_t t   = idx >> 9;
    const int    hh  = (int)(t & (HW - 1));          // H coordinate
    const size_t bc  = t >> 9;                       // b*3 + c
    const int    c   = (int)(bc % NC);
    const int    b   = (int)(bc / NC);

    const int ch = c * 256 + (hh >> 5) * 16 + (w >> 5);
    const float sv = s[(size_t)b * CKK + ch];

    v4f xv = ((const v4f*)x)[i4];
    v4f ov = xv * xv * sv;
    ((v4f*)out)[i4] = ov;
}

// ---------------------------------------------------------------------------
// Launch: pool -> gemm1 -> gemm2 -> modulate (stream order gives the syncs)
// ---------------------------------------------------------------------------
extern "C" void kernel_launch(void* const* d_in, const int* in_sizes, int n_in,
                              void* d_out, int out_size, void* d_ws, size_t ws_size,
                              hipStream_t stream) {
    const float* x  = (const float*)d_in[0];   // (16,3,512,512)
    const float* W2 = (const float*)d_in[1];   // (1536,768)
    const float* W3 = (const float*)d_in[2];   // (768,1536)
    const float* b3 = (const float*)d_in[3];   // (768,)
    float* out = (float*)d_out;

    float* pooled = (float*)d_ws;              // 16*768  floats (48 KB)
    float* hbuf   = pooled + NB * CKK;         // 16*1536 floats (96 KB)
    float* sbuf   = hbuf + NB * CHID;          // 16*768  floats (48 KB)

    // 12288 tiles, 8 waves (tiles) per 256-thread block
    se_pool<<<12288 / 8, 256, 0, stream>>>(x, pooled);
    // 96 N-tiles, 8 waves per block
    se_gemm1<<<96 / 8, 256, 0, stream>>>(pooled, W2, hbuf);
    // 48 N-tiles
    se_gemm2<<<48 / 8, 256, 0, stream>>>(hbuf, W3, b3, sbuf);
    // 16*3*512*512/4 = 3,145,728 float4s
    se_modulate<<<3145728 / 256, 256, 0, stream>>>(x, sbuf, out);
}